// Curvature_1125281431627
// MI455X (gfx1250) — compile-verified
//
#include <hip/hip_runtime.h>
#include <stdint.h>

// Curvature / TV-divergence stencil for u: (B=16, H=1024, W=1024, 1) fp32.
// Memory-bound (~128 MiB traffic -> ~5.6us floor @ 23.3 TB/s). Strategy:
// stage halo tiles into LDS with gfx1250 async global->LDS loads (ASYNCcnt),
// compute with v_rsq_f32, stream coalesced stores.

#define TX 256              // tile width  (threads per block)
#define TY 8                // tile height (outputs per thread)
#define LP (TX + 2)         // LDS row pitch in floats (258)
#define LROWS (TY + 2)      // halo rows (10)
#define NLOAD (LP * LROWS)  // 2580 floats ~= 10.3 KB LDS
#define EPSF 1e-16f

// reflect (mode='reflect', no edge repeat): -1 -> 1, n -> n-2. Halo is +/-1 only.
__device__ __forceinline__ int reflect_idx(int i, int n) {
    if (i < 0) i = -i;
    if (i >= n) i = 2 * n - 2 - i;
    return i;
}

// Async copy of one fp32 from global memory directly into LDS (no VGPR round
// trip). Tracked by ASYNCcnt. Builtin when the toolchain exposes it (param
// types are addrspace-qualified int* per the clang diagnostic), else raw
// gfx1250 ISA (VGLOBAL opcode 96, GV mode: vdst = LDS byte offset VGPR,
// vaddr = 64-bit global address VGPR pair, SADDR = off).
__device__ __forceinline__ void async_load_f32(const float* g, float* l) {
#if defined(__gfx1250__) && __has_builtin(__builtin_amdgcn_global_load_async_to_lds_b32)
    __builtin_amdgcn_global_load_async_to_lds_b32(
        (__attribute__((address_space(1))) int*)(uintptr_t)g,
        (__attribute__((address_space(3))) int*)(uint32_t)(uintptr_t)l,
        0, 0);
#else
    // Generic LDS address: low 32 bits are the wave-relative LDS offset
    // (ISA 10.2 aperture rules), which is exactly what VDST expects.
    uint32_t lds_off = (uint32_t)(uintptr_t)l;
    asm volatile("global_load_async_to_lds_b32 %0, %1, off"
                 :: "v"(lds_off), "v"(g)
                 : "memory");
#endif
}

__device__ __forceinline__ void wait_async_zero() {
#if defined(__gfx1250__) && __has_builtin(__builtin_amdgcn_s_wait_asynccnt)
    __builtin_amdgcn_s_wait_asynccnt(0);
#else
    asm volatile("s_wait_asynccnt 0x0" ::: "memory");
#endif
}

__global__ void __launch_bounds__(TX)
curvature_kernel(const float* __restrict__ u, float* __restrict__ out,
                 int H, int W) {
    __shared__ float tile[NLOAD];

    const int tid  = threadIdx.x;
    const int col0 = blockIdx.x * TX;
    const int row0 = blockIdx.y * TY;
    const size_t ibase = (size_t)blockIdx.z * (size_t)H * (size_t)W;

    // ---- Stage (TY+2) x (TX+2) halo tile into LDS via async loads ----
    for (int i = tid; i < NLOAD; i += TX) {
        const int ly = i / LP;
        const int lx = i - ly * LP;
        const int gr = reflect_idx(row0 - 1 + ly, H);
        const int gc = reflect_idx(col0 - 1 + lx, W);
        async_load_f32(u + ibase + (size_t)gr * (size_t)W + (size_t)gc,
                       &tile[i]);
    }
    wait_async_zero();   // my wave's async LDS writes are visible
    __syncthreads();     // everyone else's too

    // ---- Compute: each thread owns one column, TY rows ----
    const int lx = tid + 1;
    const int oc = col0 + tid;
#pragma unroll
    for (int ly = 1; ly <= TY; ++ly) {
        const float c  = tile[ly * LP + lx];          // u[i  , j  ]
        const float xm = tile[(ly - 1) * LP + lx];    // u[i-1, j  ]
        const float xp = tile[(ly + 1) * LP + lx];    // u[i+1, j  ]
        const float ym = tile[ly * LP + lx - 1];      // u[i  , j-1]
        const float yp = tile[ly * LP + lx + 1];      // u[i  , j+1]
        const float ne = tile[(ly - 1) * LP + lx + 1];// u[i-1, j+1]
        const float sw = tile[(ly + 1) * LP + lx - 1];// u[i+1, j-1]

        const float dxf  = xp - c;
        const float dxb  = c - xm;
        const float dyf  = yp - c;
        const float dyb  = c - ym;
        const float dsbf = ne - xm;   // forward-y diff on row i-1
        const float dslf = sw - ym;   // forward-x diff on col j-1

        const float rF = __builtin_amdgcn_rsqf(dxf * dxf + dyf * dyf + EPSF);
        const float rG = __builtin_amdgcn_rsqf(dxb * dxb + dsbf * dsbf + EPSF);
        const float rH = __builtin_amdgcn_rsqf(dslf * dslf + dyb * dyb + EPSF);

        out[ibase + (size_t)(row0 + ly - 1) * (size_t)W + (size_t)oc] =
            (dxf + dyf) * rF - dxb * rG - dyb * rH;
    }
}

extern "C" void kernel_launch(void* const* d_in, const int* in_sizes, int n_in,
                              void* d_out, int out_size, void* d_ws, size_t ws_size,
                              hipStream_t stream) {
    (void)n_in; (void)out_size; (void)d_ws; (void)ws_size;
    const float* u = (const float*)d_in[0];
    float* out = (float*)d_out;

    const int H = 1024, W = 1024;
    const int B = in_sizes[0] / (H * W);  // 16

    dim3 grid(W / TX, H / TY, B);         // (4, 128, 16)
    curvature_kernel<<<grid, TX, 0, stream>>>(u, out, H, W);
}